// GAT_77386720739612
// MI455X (gfx1250) — compile-verified
//
#include <hip/hip_runtime.h>
#include <math.h>

#define N_NODES 50000
#define E_EDGES 800000
#define E_TOT   (E_EDGES + N_NODES)
#define D_IN    128
#define HDIM    96
#define D_OUT   16
#define SLOPE   0.2f
#define BN_EPS  1e-5f

typedef float v2f __attribute__((ext_vector_type(2)));
typedef float v8f __attribute__((ext_vector_type(8)));

// ---------------------------------------------------------------------------
// WMMA fp32 GEMM:  C[M x NC] = A[M x K] * W[NC x K]^T  (+ bias) (+ relu)
// One wave32 computes one 16x16 output tile with V_WMMA_F32_16X16X4_F32.
// M, NC multiples of 16; K multiple of 4.  EPI: 0=none, 1=bias, 2=bias+relu
// ---------------------------------------------------------------------------
template <int EPI>
__global__ __launch_bounds__(256) void gemm_wmma_f32(
    const float* __restrict__ A, const float* __restrict__ W,
    const float* __restrict__ bias, float* __restrict__ C,
    int M, int K, int NC) {
  const int lane = threadIdx.x & 31;
  const int wave = threadIdx.x >> 5;
  const int nct  = NC >> 4;
  const int tile = blockIdx.x * (blockDim.x >> 5) + wave;
  if (tile >= (M >> 4) * nct) return;           // wave-uniform guard (EXEC all-1s at WMMA)
  const int rt = tile / nct;
  const int ct = tile % nct;
  const int r  = lane & 15;                     // row within 16-fragment
  const int kh = lane >> 4;                     // K pair select: 0 -> K0/K1, 1 -> K2/K3
  const float* arow = A + (size_t)(rt * 16 + r) * K + 2 * kh;
  const float* wrow = W + (size_t)(ct * 16 + r) * K + 2 * kh;
  v8f acc = {};
  for (int k = 0; k < K; k += 4) {
    v2f av = *(const v2f*)(arow + k);           // A: VGPR0=K{0|2}, VGPR1=K{1|3}
    v2f bv = *(const v2f*)(wrow + k);           // B[k][n] = W[n][k]
    acc = __builtin_amdgcn_wmma_f32_16x16x4_f32(
        false, av, false, bv, (short)0, acc, false, false);
  }
  const int cc = ct * 16 + (lane & 15);
  const float b = (EPI > 0) ? bias[cc] : 0.0f;
#pragma unroll
  for (int v = 0; v < 8; ++v) {                 // C/D: lane<16 -> M=v, lane>=16 -> M=v+8
    const int row = rt * 16 + v + 8 * kh;
    float val = acc[v] + b;
    if (EPI == 2) val = fmaxf(val, 0.0f);
    C[(size_t)row * NC + cc] = val;
  }
}

// ---------------------------------------------------------------------------
// Per-node attention scores: ssrc[n] = h[n,:].a_src, sdst[n] = h[n,:].a_dst
// ---------------------------------------------------------------------------
__global__ __launch_bounds__(256) void att_scores(
    const float* __restrict__ h, const float* __restrict__ a_src,
    const float* __restrict__ a_dst, float* __restrict__ ssrc,
    float* __restrict__ sdst, int dout) {
  const int node = blockIdx.x * (blockDim.x >> 5) + (threadIdx.x >> 5);
  if (node >= N_NODES) return;                  // wave-uniform
  const int lane = threadIdx.x & 31;
  const float* hr = h + (size_t)node * dout;
  float s1 = 0.f, s2 = 0.f;
  for (int j = lane; j < dout; j += 32) {
    const float v = hr[j];
    s1 += v * a_src[j];
    s2 += v * a_dst[j];
  }
  for (int off = 16; off; off >>= 1) {
    s1 += __shfl_xor(s1, off, 32);
    s2 += __shfl_xor(s2, off, 32);
  }
  if (lane == 0) { ssrc[node] = s1; sdst[node] = s2; }
}

// ---------------------------------------------------------------------------
// CSR-by-dst construction (built once per launch, reused by all 4 layers)
// ---------------------------------------------------------------------------
__device__ __forceinline__ void edge_sd(const int* __restrict__ ei, int e,
                                        int& s, int& d) {
  if (e < E_EDGES) { s = ei[e]; d = ei[E_EDGES + e]; }
  else             { s = d = e - E_EDGES; }       // appended self-loops
}

__global__ void deg_zero(int* __restrict__ deg) {
  const int i = blockIdx.x * blockDim.x + threadIdx.x;
  if (i < N_NODES) deg[i] = 0;
}

__global__ void deg_count(const int* __restrict__ ei, int* __restrict__ deg) {
  const int e = blockIdx.x * blockDim.x + threadIdx.x;
  if (e >= E_TOT) return;
  int s, d; edge_sd(ei, e, s, d);
  atomicAdd(&deg[d], 1);
}

// Single-block exclusive scan: rowptr[0..N], plus a scatter cursor copy.
__global__ __launch_bounds__(1024) void build_rowptr(
    const int* __restrict__ deg, int* __restrict__ rowptr,
    int* __restrict__ cursor) {
  const int T = 1024;
  const int chunk = (N_NODES + T - 1) / T;
  const int t = threadIdx.x;
  const int b = t * chunk;
  const int e = min(b + chunk, N_NODES);
  __shared__ int sums[1024];
  int s = 0;
  for (int i = b; i < e; ++i) s += deg[i];
  sums[t] = s;
  __syncthreads();
  for (int off = 1; off < T; off <<= 1) {       // inclusive Hillis-Steele scan
    const int mine = sums[t];
    const int prev = (t >= off) ? sums[t - off] : 0;
    __syncthreads();
    sums[t] = mine + prev;
    __syncthreads();
  }
  int run = (t > 0) ? sums[t - 1] : 0;          // exclusive prefix of this chunk
  for (int i = b; i < e; ++i) {
    rowptr[i] = run;
    cursor[i] = run;
    run += deg[i];
  }
  if (t == T - 1) rowptr[N_NODES] = run;        // == E_TOT
}

__global__ void csr_fill(const int* __restrict__ ei, int* __restrict__ cursor,
                         int* __restrict__ csr_src) {
  const int e = blockIdx.x * blockDim.x + threadIdx.x;
  if (e >= E_TOT) return;
  int s, d; edge_sd(ei, e, s, d);
  const int p = atomicAdd(&cursor[d], 1);
  csr_src[p] = s;
}

// ---------------------------------------------------------------------------
// Gather-based edge softmax + aggregation: one wave32 per dst node.
// Pass 1: lane-parallel segment max + exp-sum (shfl_xor trees).
// Pass 2: per-edge loop, lanes read 96 consecutive floats of h[src]
//         (coalesced, L2-resident) and accumulate in registers. No atomics.
// ---------------------------------------------------------------------------
__global__ __launch_bounds__(256) void gat_aggregate(
    const int* __restrict__ rowptr, const int* __restrict__ csr_src,
    const float* __restrict__ ssrc, const float* __restrict__ sdst,
    const float* __restrict__ h, float* __restrict__ out, int dout) {
  const int node = blockIdx.x * (blockDim.x >> 5) + (threadIdx.x >> 5);
  if (node >= N_NODES) return;                  // wave-uniform
  const int lane = threadIdx.x & 31;
  const int beg = rowptr[node];
  const int end = rowptr[node + 1];
  const float sd = sdst[node];

  // segment max (stable softmax)
  float mx = -3.0e38f;
  for (int i = beg + lane; i < end; i += 32) {
    float v = ssrc[csr_src[i]] + sd;
    v = (v >= 0.f) ? v : SLOPE * v;
    mx = fmaxf(mx, v);
  }
  for (int off = 16; off; off >>= 1) mx = fmaxf(mx, __shfl_xor(mx, off, 32));

  // segment exp-sum
  float sm = 0.f;
  for (int i = beg + lane; i < end; i += 32) {
    float v = ssrc[csr_src[i]] + sd;
    v = (v >= 0.f) ? v : SLOPE * v;
    sm += expf(v - mx);
  }
  for (int off = 16; off; off >>= 1) sm += __shfl_xor(sm, off, 32);
  const float inv = 1.0f / sm;

  // weighted feature gather (dout == 96: 3 registers per lane)
  float a0 = 0.f, a1 = 0.f, a2 = 0.f;
  for (int i = beg; i < end; ++i) {
    const int s = csr_src[i];                   // uniform per wave -> broadcast
    float v = ssrc[s] + sd;
    v = (v >= 0.f) ? v : SLOPE * v;
    const float alpha = expf(v - mx) * inv;
    const float* hr = h + (size_t)s * dout;
    a0 += alpha * hr[lane];
    a1 += alpha * hr[lane + 32];
    a2 += alpha * hr[lane + 64];
  }
  float* o = out + (size_t)node * dout;
  o[lane]      = a0;
  o[lane + 32] = a1;
  o[lane + 64] = a2;
}

// Column-wise mean / biased var over N rows (one block per column)
__global__ __launch_bounds__(256) void bn_stats(const float* __restrict__ h,
                                                float* __restrict__ mean,
                                                float* __restrict__ var, int dout) {
  const int c = blockIdx.x;
  float s = 0.f, s2 = 0.f;
  for (int r = threadIdx.x; r < N_NODES; r += blockDim.x) {
    const float v = h[(size_t)r * dout + c];
    s += v; s2 += v * v;
  }
  __shared__ float sh[256], sh2[256];
  sh[threadIdx.x] = s; sh2[threadIdx.x] = s2;
  __syncthreads();
  for (int off = blockDim.x >> 1; off; off >>= 1) {
    if (threadIdx.x < off) {
      sh[threadIdx.x]  += sh[threadIdx.x + off];
      sh2[threadIdx.x] += sh2[threadIdx.x + off];
    }
    __syncthreads();
  }
  if (threadIdx.x == 0) {
    const float mu = sh[0] / (float)N_NODES;
    mean[c] = mu;
    var[c]  = sh2[0] / (float)N_NODES - mu * mu;
  }
}

__global__ void bn_relu(const float* __restrict__ h, const float* __restrict__ mean,
                        const float* __restrict__ var, const float* __restrict__ gamma,
                        const float* __restrict__ beta, float* __restrict__ out,
                        int dout) {
  const int i = blockIdx.x * blockDim.x + threadIdx.x;
  if (i >= N_NODES * dout) return;
  const int c = i % dout;
  float v = (h[i] - mean[c]) * rsqrtf(var[c] + BN_EPS);
  v = gamma[c] * v + beta[c];
  out[i] = fmaxf(v, 0.0f);
}

// ---------------------------------------------------------------------------
static inline int cdiv(long long a, long long b) { return (int)((a + b - 1) / b); }

extern "C" void kernel_launch(void* const* d_in, const int* in_sizes, int n_in,
                              void* d_out, int out_size, void* d_ws, size_t ws_size,
                              hipStream_t stream) {
  const float* x  = (const float*)d_in[0];
  const int*   ei = (const int*)d_in[1];
  const float* Ws[4]   = {(const float*)d_in[2],  (const float*)d_in[3],
                          (const float*)d_in[4],  (const float*)d_in[5]};
  const float* asrc[4] = {(const float*)d_in[6],  (const float*)d_in[7],
                          (const float*)d_in[8],  (const float*)d_in[9]};
  const float* adst[4] = {(const float*)d_in[10], (const float*)d_in[11],
                          (const float*)d_in[12], (const float*)d_in[13]};
  // d_in[14..17] = gat_biases (zero, and provably cancel inside BatchNorm)
  const float* gam[4]  = {(const float*)d_in[18], (const float*)d_in[19],
                          (const float*)d_in[20], (const float*)d_in[21]};
  const float* bet[4]  = {(const float*)d_in[22], (const float*)d_in[23],
                          (const float*)d_in[24], (const float*)d_in[25]};
  const float* lin1_w = (const float*)d_in[26];
  const float* lin1_b = (const float*)d_in[27];
  const float* lin2_w = (const float*)d_in[28];
  const float* lin2_b = (const float*)d_in[29];

  // Workspace carve-up
  char* wsb = (char*)d_ws;
  size_t off = 0;
  float* hW    = (float*)(wsb + off); off += (size_t)N_NODES * HDIM * 4;
  float* hAgg  = (float*)(wsb + off); off += (size_t)N_NODES * HDIM * 4;
  float* hNorm = (float*)(wsb + off); off += (size_t)N_NODES * HDIM * 4;
  float* ssrc  = (float*)(wsb + off); off += (size_t)N_NODES * 4;
  float* sdst  = (float*)(wsb + off); off += (size_t)N_NODES * 4;
  float* mean  = (float*)(wsb + off); off += 128 * 4;
  float* var   = (float*)(wsb + off); off += 128 * 4;
  int* deg     = (int*)(wsb + off);   off += (size_t)N_NODES * 4;
  int* rowptr  = (int*)(wsb + off);   off += (size_t)(N_NODES + 1) * 4;
  int* cursor  = (int*)(wsb + off);   off += (size_t)N_NODES * 4;
  int* csr_src = (int*)(wsb + off);   off += (size_t)E_TOT * 4;

  const int TPB = 256;

  // ---- Build dst-CSR once (in-row order set by atomics; reused by all 4 layers) ----
  deg_zero<<<cdiv(N_NODES, TPB), TPB, 0, stream>>>(deg);
  deg_count<<<cdiv(E_TOT, TPB), TPB, 0, stream>>>(ei, deg);
  build_rowptr<<<1, 1024, 0, stream>>>(deg, rowptr, cursor);
  csr_fill<<<cdiv(E_TOT, TPB), TPB, 0, stream>>>(ei, cursor, csr_src);

  // ---- 4 GAT layers ----
  const float* hin = x;
  int din = D_IN;
  for (int l = 0; l < 4; ++l) {
    const int tiles = (N_NODES / 16) * (HDIM / 16);
    gemm_wmma_f32<0><<<cdiv(tiles, 8), TPB, 0, stream>>>(
        hin, Ws[l], nullptr, hW, N_NODES, din, HDIM);
    att_scores<<<cdiv(N_NODES, 8), TPB, 0, stream>>>(
        hW, asrc[l], adst[l], ssrc, sdst, HDIM);
    gat_aggregate<<<cdiv(N_NODES, 8), TPB, 0, stream>>>(
        rowptr, csr_src, ssrc, sdst, hW, hAgg, HDIM);
    bn_stats<<<HDIM, TPB, 0, stream>>>(hAgg, mean, var, HDIM);
    bn_relu<<<cdiv((long long)N_NODES * HDIM, TPB), TPB, 0, stream>>>(
        hAgg, mean, var, gam[l], bet[l], hNorm, HDIM);
    hin = hNorm;
    din = HDIM;
  }

  // ---- MLP head: lin1 applied twice (faithful to reference), then lin2 ----
  const int t96 = (N_NODES / 16) * (HDIM / 16);
  gemm_wmma_f32<2><<<cdiv(t96, 8), TPB, 0, stream>>>(
      hNorm, lin1_w, lin1_b, hW, N_NODES, HDIM, HDIM);
  gemm_wmma_f32<2><<<cdiv(t96, 8), TPB, 0, stream>>>(
      hW, lin1_w, lin1_b, hAgg, N_NODES, HDIM, HDIM);
  const int t16 = (N_NODES / 16) * (D_OUT / 16);
  gemm_wmma_f32<1><<<cdiv(t16, 8), TPB, 0, stream>>>(
      hAgg, lin2_w, lin2_b, (float*)d_out, N_NODES, HDIM, D_OUT);
}